// Qfnn_29420525977860
// MI455X (gfx1250) — compile-verified
//
#include <hip/hip_runtime.h>

// ---------------------------------------------------------------------------
// Qfnn forward: the reference output is broadcast_to(pv, (B, 3)) where pv
// depends ONLY on theta_d (input 12); the whole NN pipeline is dead code.
//
// The parameter-shift average collapses analytically:
//   avg over ±pi/2 shifts of cos^2/sin^2((t±pi/2)/2) = 1/2, so with
//   C_j=(1+cos t_j)/2, S_j=(1-cos t_j)/2:
//     pv0 = (C1C2 + C0C2 + C0C1)/6
//     pv1 = (C1C2 + S0C2 + S0C1)/6
//     pv2 = (S1C2 + C0C2 + C0S1)/6
// -> only 3 cosines, then a 3 MiB nontemporal broadcast-store fill
//    (store-bandwidth bound: 3 MiB / 23.3 TB/s ~ 135 ns).
// ---------------------------------------------------------------------------

typedef float v4f __attribute__((ext_vector_type(4)));

// Compact accurate cosf: Cody-Waite pi/2 reduction + cephes minimax polys.
// Accurate to ~1e-7 for |x| < ~1e4 (theta_d ~ N(0,1)); no libm slow path.
__device__ __forceinline__ float qfnn_cos(float x) {
    const float n = rintf(x * 0.63661977236758134f);       // x * 2/pi
    float r = __builtin_fmaf(n, -1.57079637050629e+0f, x); // - n * PIO2_HI
    r = __builtin_fmaf(n, 4.37113882867379e-8f, r);        // - n * PIO2_LO
    const float z = r * r;
    // sin(r), cos(r) on [-pi/4, pi/4]
    const float sp = r * __builtin_fmaf(
        z, __builtin_fmaf(z, __builtin_fmaf(z, -1.9515295891e-4f,
                                               8.3321608736e-3f),
                             -1.6666654611e-1f),
        1.0f);
    const float cp = __builtin_fmaf(
        z * z, __builtin_fmaf(z, __builtin_fmaf(z, 2.443315711809948e-5f,
                                                  -1.388731625493765e-3f),
                                 4.166664568298827e-2f),
        __builtin_fmaf(z, -0.5f, 1.0f));
    const int q = ((int)n) & 3;  // two's complement & handles negative n
    // cos(n*pi/2 + r): q==0 -> cos r, 1 -> -sin r, 2 -> -cos r, 3 -> sin r
    const float mag = (q & 1) ? sp : cp;
    return ((q == 1) | (q == 2)) ? -mag : mag;
}

__global__ __launch_bounds__(256) void qfnn_broadcast_fill(
    const float* __restrict__ theta_d,
    float* __restrict__ out,
    int n_elems)  // total floats = B * 3
{
    // Wave-uniform pv computation (compiler scalarizes; ~80 ops total).
    const float k0 = qfnn_cos(theta_d[0]);
    const float k1 = qfnn_cos(theta_d[1]);
    const float k2 = qfnn_cos(theta_d[2]);
    const float C0 = 0.5f * (1.0f + k0), S0 = 0.5f * (1.0f - k0);
    const float C1 = 0.5f * (1.0f + k1), S1 = 0.5f * (1.0f - k1);
    const float C2 = 0.5f * (1.0f + k2);
    const float SIXTH = 1.0f / 6.0f;
    const float p0 = (C1 * C2 + C0 * C2 + C0 * C1) * SIXTH;
    const float p1 = (C1 * C2 + S0 * C2 + S0 * C1) * SIXTH;
    const float p2 = (S1 * C2 + C0 * C2 + C0 * S1) * SIXTH;

    // Output pattern repeats every 3 floats; a float4 at index t holds
    // pv[(4t+u) mod 3] -> 3 distinct float4 patterns, keyed by t % 3.
    const v4f pat0 = { p0, p1, p2, p0 };
    const v4f pat1 = { p1, p2, p0, p1 };
    const v4f pat2 = { p2, p0, p1, p2 };

    v4f* __restrict__ out4 = reinterpret_cast<v4f*>(out);
    const int n4     = n_elems >> 2;
    const int stride = (int)(gridDim.x * blockDim.x);
    const int gid    = (int)(blockIdx.x * blockDim.x + threadIdx.x);

    for (int t = gid; t < n4; t += stride) {
        const int r = t % 3;
        const v4f v = (r == 0) ? pat0 : ((r == 1) ? pat1 : pat2);
        __builtin_nontemporal_store(v, out4 + t);  // global_store_b128, NT
    }

    // Tail guard (never hit for B*3 = 786432, but safe for any size).
    const int tail = n_elems & 3;
    if (gid < tail) {
        const int idx = (n4 << 2) + gid;
        const int r = idx % 3;
        out[idx] = (r == 0) ? p0 : ((r == 1) ? p1 : p2);
    }
}

extern "C" void kernel_launch(void* const* d_in, const int* in_sizes, int n_in,
                              void* d_out, int out_size, void* d_ws, size_t ws_size,
                              hipStream_t stream) {
    (void)in_sizes; (void)n_in; (void)d_ws; (void)ws_size;

    // setup_inputs() order: x, W1, b1, gn_w, gn_b, m, theta_f, theta_t,
    //                       bn_w, bn_b, W2, b2, theta_d  -> theta_d = d_in[12]
    const float* theta_d = (const float*)d_in[12];
    float* out = (float*)d_out;

    const int block = 256;             // 8 wave32 waves per block
    const int n4 = out_size >> 2;      // 196608 float4 stores
    int grid = (n4 + block * 2 - 1) / (block * 2);  // ~2 float4 / thread
    if (grid < 1) grid = 1;            // -> 384 blocks for B = 262144

    qfnn_broadcast_fill<<<grid, block, 0, stream>>>(theta_d, out, out_size);
}